// IPEXQKVFusedGemm_47407849013384
// MI455X (gfx1250) — compile-verified
//
#include <hip/hip_runtime.h>

// fp32 WMMA fragment types (wave32): A/B 16x4 & 4x16 f32 = 2 VGPRs, C/D 16x16 f32 = 8 VGPRs
typedef __attribute__((ext_vector_type(2))) float v2f;
typedef __attribute__((ext_vector_type(8))) float v8f;
typedef __attribute__((ext_vector_type(4))) int   v4i;

#define TOKENS    64
#define EMBED     4096                     // K
#define NKV       8
#define QKV_GROUP 6
#define HDIM      128
#define NCOLS     (NKV * QKV_GROUP * HDIM) // 6144 = N
#define QWIDTH    4096                     // 32 heads * 128
#define KVWIDTH   1024                     // 8 kv heads * 128
#define KC        64                       // K chunk staged in LDS
#define KCP       68                       // padded A row stride (dwords): conflict-free frag loads
#define NLOC      64                       // N columns per block
#define SB        72                       // padded B row stride (dwords): 8-bank rotation per k-row
#define WAVES_PB  4

// ---- CDNA5 async global->LDS path (ASYNCcnt), with portable fallback ----
#if defined(__has_builtin)
# if __has_builtin(__builtin_amdgcn_global_load_async_to_lds_b128)
#  define HAVE_ASYNC_LDS 1
# endif
#endif
#ifndef HAVE_ASYNC_LDS
# define HAVE_ASYNC_LDS 0
#endif

typedef __attribute__((address_space(1))) v4i gas_v4i;  // global int4
typedef __attribute__((address_space(3))) v4i las_v4i;  // LDS int4

__device__ __forceinline__ void copy_b128_to_lds(const float* g, float* l) {
#if HAVE_ASYNC_LDS
    __builtin_amdgcn_global_load_async_to_lds_b128(
        (gas_v4i*)g, (las_v4i*)l, 0, 0);
#else
    *reinterpret_cast<float4*>(l) = *reinterpret_cast<const float4*>(g);
#endif
}

__device__ __forceinline__ void async_wait_all() {
#if HAVE_ASYNC_LDS
# if __has_builtin(__builtin_amdgcn_s_wait_asynccnt)
    __builtin_amdgcn_s_wait_asynccnt(0);
# else
    asm volatile("s_wait_asynccnt 0x0" ::: "memory");
# endif
#endif
}

__global__ __launch_bounds__(WAVES_PB * 32)
void qkv_fused_wmma_f32(const float* __restrict__ inp,
                        const float* __restrict__ w,
                        const float* __restrict__ bias,
                        float* __restrict__ out)
{
    // double-buffered chunks: A = 64 x 68 dwords, B = 64 x 72 dwords (~72 KB total)
    __shared__ __align__(16) float As[2][TOKENS * KCP];
    __shared__ __align__(16) float Bs[2][KC * SB];

    const int tid    = threadIdx.x;
    const int wave   = tid >> 5;
    const int lane   = tid & 31;
    const int laneN  = lane & 15;   // N (or M) index within fragment
    const int laneHi = lane >> 4;   // selects K pair {0,1} vs {2,3}

    const int nbase = blockIdx.x * NLOC;        // block's first output column
    const int col   = nbase + wave * 16 + laneN;
    const int bcol  = wave * 16 + laneN;        // column within Bs chunk

    v8f acc0 = {}, acc1 = {}, acc2 = {}, acc3 = {};

    // stage one K chunk (A: 64x64, B: 64x64) into LDS buffer `buf`
    auto stage = [&](int kc, int buf) {
        #pragma unroll
        for (int j = 0; j < 8; ++j) {           // A: 1024 float4, coalesced
            int idx4 = tid + j * 128;
            int row  = idx4 >> 4;               // 16 float4 per 64-col row
            int k4   = (idx4 & 15) << 2;
            copy_b128_to_lds(inp + (size_t)row * EMBED + kc + k4,
                             &As[buf][row * KCP + k4]);
        }
        #pragma unroll
        for (int j = 0; j < 8; ++j) {           // B: 1024 float4, coalesced
            int idx4 = tid + j * 128;
            int k    = idx4 >> 4;
            int n4   = (idx4 & 15) << 2;
            copy_b128_to_lds(w + (size_t)(kc + k) * NCOLS + nbase + n4,
                             &Bs[buf][k * SB + n4]);
        }
    };

    stage(0, 0);
    async_wait_all();
    __syncthreads();

    const int NCHUNK = EMBED / KC;              // 64
    for (int c = 0; c < NCHUNK; ++c) {
        const int buf = c & 1;
        if (c + 1 < NCHUNK) stage((c + 1) * KC, buf ^ 1);  // prefetch next chunk (async)

        #pragma unroll 4
        for (int ks = 0; ks < KC; ks += 4) {
            // B fragment (4x16 f32): lanes 0-15 k={ks,ks+1}, lanes 16-31 k={ks+2,ks+3}
            const int bb = (ks + laneHi * 2) * SB + bcol;
            v2f b;
            b.x = Bs[buf][bb];
            b.y = Bs[buf][bb + SB];

            // A fragments (16x4 f32): row = mtile*16 + laneN, k pair by laneHi
            const int ao = ks + laneHi * 2;
            v2f a;
            a = *reinterpret_cast<const v2f*>(&As[buf][(laneN +  0) * KCP + ao]);
            acc0 = __builtin_amdgcn_wmma_f32_16x16x4_f32(false, a, false, b, (short)0, acc0, false, false);
            a = *reinterpret_cast<const v2f*>(&As[buf][(laneN + 16) * KCP + ao]);
            acc1 = __builtin_amdgcn_wmma_f32_16x16x4_f32(false, a, false, b, (short)0, acc1, false, false);
            a = *reinterpret_cast<const v2f*>(&As[buf][(laneN + 32) * KCP + ao]);
            acc2 = __builtin_amdgcn_wmma_f32_16x16x4_f32(false, a, false, b, (short)0, acc2, false, false);
            a = *reinterpret_cast<const v2f*>(&As[buf][(laneN + 48) * KCP + ao]);
            acc3 = __builtin_amdgcn_wmma_f32_16x16x4_f32(false, a, false, b, (short)0, acc3, false, false);
        }

        async_wait_all();   // own async copies for chunk c+1 have landed
        __syncthreads();    // make them visible; fence reuse of buffer c
    }

    // Epilogue: add bias, scatter into q / k / v regions of d_out.
    const float bv  = bias[col];
    const int   kv  = col / (QKV_GROUP * HDIM);          // 768 cols per kv head
    const int   rem = col - kv * (QKV_GROUP * HDIM);
    const int   g   = rem / HDIM;
    const int   d   = rem - g * HDIM;

    size_t base;
    int    stride, colOff;
    if (g < QKV_GROUP - 2) {                             // q: [T, 32*128]
        base   = 0;
        stride = QWIDTH;
        colOff = kv * ((QKV_GROUP - 2) * HDIM) + g * HDIM + d;
    } else {                                             // k or v: [T, 8*128]
        stride = KVWIDTH;
        base   = (size_t)TOKENS * QWIDTH + (size_t)(g - (QKV_GROUP - 2)) * TOKENS * KVWIDTH;
        colOff = kv * HDIM + d;
    }
    float* op = out + base + colOff;

    // C/D layout: VGPR r holds M = r (lanes 0-15) or M = 8+r (lanes 16-31)
    #pragma unroll
    for (int r = 0; r < 8; ++r) {
        const int t0 = r + 8 * laneHi;
        op[(size_t)(t0 +  0) * stride] = acc0[r] + bv;
        op[(size_t)(t0 + 16) * stride] = acc1[r] + bv;
        op[(size_t)(t0 + 32) * stride] = acc2[r] + bv;
        op[(size_t)(t0 + 48) * stride] = acc3[r] + bv;
    }
}

extern "C" void kernel_launch(void* const* d_in, const int* in_sizes, int n_in,
                              void* d_out, int out_size, void* d_ws, size_t ws_size,
                              hipStream_t stream) {
    const float* inp  = (const float*)d_in[0];  // [64, 4096]
    const float* w    = (const float*)d_in[1];  // [4096, 8, 6, 128] = [4096, 6144]
    const float* bias = (const float*)d_in[2];  // [8, 6, 128] = [6144]
    float* out = (float*)d_out;                 // q(64*4096) | k(64*1024) | v(64*1024)

    dim3 grid(NCOLS / NLOC);    // 96 blocks
    dim3 block(WAVES_PB * 32);  // 128 threads = 4 waves
    qkv_fused_wmma_f32<<<grid, block, 0, stream>>>(inp, w, bias, out);
}